// ManualMaxPool2DLayer_1099511627946
// MI455X (gfx1250) — compile-verified
//
#include <hip/hip_runtime.h>
#include <hip/hip_bf16.h>

// Max-pool 2x2 stride 2, VALID, NCHW f32.
// x: (64, 96, 224, 224) -> out: (64, 96, 112, 112)
//
// Memory-bound: 1.232 GB read + 0.308 GB written once each -> ~66 us floor at
// 23.3 TB/s. Each thread: one float4 load from each of two adjacent input rows ->
// 2 outputs (float2 store). Non-temporal hints keep the 1.5 GB stream from
// rinsing the 192 MB L2.
//
// Indexing is folded into launch geometry (no integer division anywhere):
//   threadIdx.x        = w2    in [0,56)  : which output pair in the row
//   blockIdx.y*4+t.y   = oh    in [0,112) : output row
//   blockIdx.z         = plane in [0,6144): N*C plane
// Block = 56*4 = 224 threads = exactly 7 wave32s.

typedef float v4f __attribute__((ext_vector_type(4)));
typedef float v2f __attribute__((ext_vector_type(2)));

#define IN_W      224
#define OUT_W     112
#define OUT_H     112
#define PLANES    (64 * 96)            // 6144
#define IN_PLANE  (224 * 224)          // 50176
#define OUT_PLANE (OUT_H * OUT_W)      // 12544
#define ROWS_PER_BLOCK 4

__global__ __launch_bounds__(56 * ROWS_PER_BLOCK) void
maxpool2x2_s2_kernel(const float* __restrict__ x, float* __restrict__ out)
{
    const unsigned int w2    = threadIdx.x;                              // 0..55
    const unsigned int oh    = blockIdx.y * ROWS_PER_BLOCK + threadIdx.y; // 0..111
    const unsigned int plane = blockIdx.z;                               // 0..6143

    // Input: rows 2*oh, 2*oh+1, cols [4*w2, 4*w2+4). 16B-aligned.
    const size_t in_base = (size_t)plane * IN_PLANE
                         + (size_t)(oh * 2u) * IN_W
                         + (size_t)(w2 * 4u);

    v4f r0 = __builtin_nontemporal_load((const v4f*)(x + in_base));        // b128 NT
    v4f r1 = __builtin_nontemporal_load((const v4f*)(x + in_base + IN_W)); // b128 NT

    // Two 2x2 windows: {r0.xy, r1.xy} and {r0.zw, r1.zw}.
    v2f o;
    o.x = fmaxf(fmaxf(r0.x, r0.y), fmaxf(r1.x, r1.y));
    o.y = fmaxf(fmaxf(r0.z, r0.w), fmaxf(r1.z, r1.w));

    const size_t out_idx = (size_t)plane * OUT_PLANE
                         + (size_t)oh * OUT_W
                         + (size_t)(w2 * 2u);
    __builtin_nontemporal_store(o, (v2f*)(out + out_idx));                  // b64 NT
}

extern "C" void kernel_launch(void* const* d_in, const int* in_sizes, int n_in,
                              void* d_out, int out_size, void* d_ws, size_t ws_size,
                              hipStream_t stream)
{
    (void)in_sizes; (void)n_in; (void)out_size; (void)d_ws; (void)ws_size;

    const float* x = (const float*)d_in[0];
    float* out = (float*)d_out;

    dim3 block(56, ROWS_PER_BLOCK, 1);                 // 224 threads = 7 waves
    dim3 grid(1, OUT_H / ROWS_PER_BLOCK, PLANES);      // (1, 28, 6144) exact cover

    maxpool2x2_s2_kernel<<<grid, block, 0, stream>>>(x, out);
}